// GCNWithDynamicLayersNumber_60584808678014
// MI455X (gfx1250) — compile-verified
//
#include <hip/hip_runtime.h>
#include <math.h>

typedef float v2f __attribute__((ext_vector_type(2)));
typedef float v8f __attribute__((ext_vector_type(8)));

#define N_NODES  50000
#define N_EDGES  600000
#define N_GRAPHS 64
#define DIM      128
#define H3DIM    384      // DIM * 3 hops
#define OUT_DIM  10

// ---------------------------------------------------------------------------
// Degree / normalization:  deg[i] = 1 + #incoming edges;  dinv = rsqrt(deg)
// ---------------------------------------------------------------------------
__global__ void deg_init_kernel(float* __restrict__ deg) {
    int i = blockIdx.x * blockDim.x + threadIdx.x;
    if (i < N_NODES) deg[i] = 1.0f;  // self loop
}

__global__ void deg_scatter_kernel(const long long* __restrict__ dst,
                                   float* __restrict__ deg) {
    int e = blockIdx.x * blockDim.x + threadIdx.x;
    if (e < N_EDGES) atomicAdd(&deg[(int)dst[e]], 1.0f);
}

__global__ void dinv_kernel(float* __restrict__ deg) {
    int i = blockIdx.x * blockDim.x + threadIdx.x;
    if (i < N_NODES) deg[i] = rsqrtf(deg[i]);
}

// ---------------------------------------------------------------------------
// Dense GEMM via V_WMMA_F32_16X16X4_F32:  out[N x 128] = A[N x 128] @ W[128 x 128]
// One wave computes one 16x16 output tile; a 256-thread block (8 waves)
// covers a 16-row strip across all 8 column tiles. N = 3125 * 16 exactly.
// ---------------------------------------------------------------------------
__global__ __launch_bounds__(256) void gemm_wmma_kernel(const float* __restrict__ A,
                                                        const float* __restrict__ W,
                                                        float* __restrict__ out) {
    const int wave    = threadIdx.x >> 5;   // 0..7 -> column tile
    const int lane    = threadIdx.x & 31;
    const int l15     = lane & 15;
    const int hi      = (lane >> 4) & 1;    // upper half-wave
    const int rowBase = blockIdx.x * 16;
    const int colBase = wave * 16;
    const int koff    = hi * 2;             // upper half holds K+2, K+3

    const float* __restrict__ arow = A + (size_t)(rowBase + l15) * DIM;

    v8f acc = {};
#pragma unroll
    for (int k = 0; k < DIM; k += 4) {
        v2f a, b;
        // A 16x4 fragment: lanes 0-15: K = k,k+1 ; lanes 16-31: K = k+2,k+3 ; M = lane&15
        a.x = arow[k + koff];
        a.y = arow[k + koff + 1];
        // B 4x16 fragment: row (K) striped across lanes; mirrored half-wave split
        b.x = W[(size_t)(k + koff) * DIM + colBase + l15];
        b.y = W[(size_t)(k + koff + 1) * DIM + colBase + l15];
        acc = __builtin_amdgcn_wmma_f32_16x16x4_f32(false, a, false, b,
                                                    (short)0, acc, false, false);
    }

    // C/D layout: VGPR v -> M = v (lanes 0-15) / M = v+8 (lanes 16-31), N = lane&15
#pragma unroll
    for (int v = 0; v < 8; ++v) {
        int m = hi * 8 + v;
        out[(size_t)(rowBase + m) * DIM + colBase + l15] = acc[v];
    }
}

// ---------------------------------------------------------------------------
// out[i,:] = xw[i,:] * dinv[i]^2 + bias   (self-loop message + bias)
// ---------------------------------------------------------------------------
__global__ void selfloop_bias_kernel(const float* __restrict__ xw,
                                     const float* __restrict__ dinv,
                                     const float* __restrict__ bias,
                                     float* __restrict__ out) {
    int idx = blockIdx.x * blockDim.x + threadIdx.x;
    if (idx >= N_NODES * DIM) return;
    int i = idx >> 7;
    int f = idx & (DIM - 1);
    float di = dinv[i];
    out[idx] = xw[idx] * di * di + bias[f];
}

// ---------------------------------------------------------------------------
// Edge scatter: one wave32 per edge, each lane moves a float4 slice.
// out[dst,:] += xw[src,:] * dinv[src] * dinv[dst]
// ---------------------------------------------------------------------------
__global__ __launch_bounds__(256) void edge_scatter_kernel(const long long* __restrict__ src,
                                                           const long long* __restrict__ dst,
                                                           const float* __restrict__ dinv,
                                                           const float* __restrict__ xw,
                                                           float* __restrict__ out) {
    int gid  = blockIdx.x * blockDim.x + threadIdx.x;
    int e    = gid >> 5;
    int lane = gid & 31;
    if (e >= N_EDGES) return;
    int s = (int)src[e];
    int d = (int)dst[e];
    float coef = dinv[s] * dinv[d];
    const float4 v = *(const float4*)(xw + (size_t)s * DIM + lane * 4);
    float* o = out + (size_t)d * DIM + lane * 4;
    atomicAdd(o + 0, v.x * coef);
    atomicAdd(o + 1, v.y * coef);
    atomicAdd(o + 2, v.z * coef);
    atomicAdd(o + 3, v.w * coef);
}

// ---------------------------------------------------------------------------
// Global mean pool over 3 layer outputs, concatenated -> pooled[64 x 384]
// ---------------------------------------------------------------------------
__global__ void pool_init_kernel(float* __restrict__ pooled, float* __restrict__ cnt) {
    int i = blockIdx.x * blockDim.x + threadIdx.x;
    if (i < N_GRAPHS * H3DIM) pooled[i] = 0.0f;
    if (i < N_GRAPHS)         cnt[i]    = 0.0f;
}

__global__ void cnt_kernel(const long long* __restrict__ batch, float* __restrict__ cnt) {
    int i = blockIdx.x * blockDim.x + threadIdx.x;
    if (i < N_NODES) atomicAdd(&cnt[(int)batch[i]], 1.0f);
}

__global__ void pool_scatter_kernel(const long long* __restrict__ batch,
                                    const float* __restrict__ h1,
                                    const float* __restrict__ h2,
                                    const float* __restrict__ h3,
                                    float* __restrict__ pooled) {
    int idx = blockIdx.x * blockDim.x + threadIdx.x;
    if (idx >= N_NODES * DIM) return;
    int node = idx >> 7;
    int f    = idx & (DIM - 1);
    int g    = (int)batch[node];
    float* pg = pooled + (size_t)g * H3DIM;
    atomicAdd(pg + f,           h1[idx]);
    atomicAdd(pg + DIM + f,     h2[idx]);
    atomicAdd(pg + 2 * DIM + f, h3[idx]);
}

__global__ void mean_kernel(float* __restrict__ pooled, const float* __restrict__ cnt) {
    int i = blockIdx.x * blockDim.x + threadIdx.x;
    if (i >= N_GRAPHS * H3DIM) return;
    pooled[i] /= fmaxf(cnt[i / H3DIM], 1.0f);
}

// ---------------------------------------------------------------------------
// Tiny MLP head (64x384 @ 384x384 -> relu -> @ 384x10 -> log_softmax)
// ---------------------------------------------------------------------------
__global__ void lin1_kernel(const float* __restrict__ pooled,
                            const float* __restrict__ W,
                            const float* __restrict__ b,
                            float* __restrict__ h) {
    int idx = blockIdx.x * blockDim.x + threadIdx.x;
    if (idx >= N_GRAPHS * H3DIM) return;
    int r = idx / H3DIM;
    int c = idx % H3DIM;
    const float* p = pooled + (size_t)r * H3DIM;
    float s = b[c];
    for (int k = 0; k < H3DIM; ++k) s += p[k] * W[(size_t)k * H3DIM + c];
    h[idx] = fmaxf(s, 0.0f);
}

__global__ void lin2_kernel(const float* __restrict__ h,
                            const float* __restrict__ W,
                            const float* __restrict__ b,
                            float* __restrict__ out) {
    int idx = blockIdx.x * blockDim.x + threadIdx.x;
    if (idx >= N_GRAPHS * OUT_DIM) return;
    int r = idx / OUT_DIM;
    int c = idx % OUT_DIM;
    const float* hr = h + (size_t)r * H3DIM;
    float s = b[c];
    for (int k = 0; k < H3DIM; ++k) s += hr[k] * W[(size_t)k * OUT_DIM + c];
    out[idx] = s;
}

__global__ void logsoftmax_kernel(float* __restrict__ out) {
    int r = blockIdx.x * blockDim.x + threadIdx.x;
    if (r >= N_GRAPHS) return;
    const float* x = out + (size_t)r * OUT_DIM;
    float m = x[0];
    for (int j = 1; j < OUT_DIM; ++j) m = fmaxf(m, x[j]);
    float s = 0.0f;
    for (int j = 0; j < OUT_DIM; ++j) s += expf(x[j] - m);
    float lse = logf(s);
    float* y = out + N_GRAPHS * OUT_DIM + (size_t)r * OUT_DIM;
    for (int j = 0; j < OUT_DIM; ++j) y[j] = x[j] - m - lse;
}

// ---------------------------------------------------------------------------
// Launch
// ---------------------------------------------------------------------------
extern "C" void kernel_launch(void* const* d_in, const int* in_sizes, int n_in,
                              void* d_out, int out_size, void* d_ws, size_t ws_size,
                              hipStream_t stream) {
    (void)in_sizes; (void)n_in; (void)out_size; (void)ws_size;

    const float* x      = (const float*)d_in[0];
    const float* W1     = (const float*)d_in[1];
    const float* b1     = (const float*)d_in[2];
    const float* W2     = (const float*)d_in[3];
    const float* b2     = (const float*)d_in[4];
    const float* W3     = (const float*)d_in[5];
    const float* b3     = (const float*)d_in[6];
    const float* lin1_W = (const float*)d_in[7];
    const float* lin1_b = (const float*)d_in[8];
    const float* lin2_W = (const float*)d_in[9];
    const float* lin2_b = (const float*)d_in[10];
    const long long* edge  = (const long long*)d_in[11];  // int64 [2, N_EDGES]
    const long long* batch = (const long long*)d_in[12];  // int64 [N_NODES]
    const long long* src = edge;
    const long long* dst = edge + N_EDGES;

    // Workspace carve-out (floats). Total ~102.8 MB — fits the 192 MB L2.
    float* ws     = (float*)d_ws;
    float* dinv   = ws;                                   // 50000 (padded 50048)
    float* xw     = dinv + 50048;                         // 6.4M, reused per layer
    float* h1     = xw + (size_t)N_NODES * DIM;           // 6.4M
    float* h2     = h1 + (size_t)N_NODES * DIM;           // 6.4M
    float* h3     = h2 + (size_t)N_NODES * DIM;           // 6.4M
    float* pooled = h3 + (size_t)N_NODES * DIM;           // 64*384
    float* cnt    = pooled + N_GRAPHS * H3DIM;            // 64
    float* hmid   = cnt + N_GRAPHS;                       // 64*384

    const int B = 256;

    // deg -> dinv
    deg_init_kernel   <<<(N_NODES + B - 1) / B, B, 0, stream>>>(dinv);
    deg_scatter_kernel<<<(N_EDGES + B - 1) / B, B, 0, stream>>>(dst, dinv);
    dinv_kernel       <<<(N_NODES + B - 1) / B, B, 0, stream>>>(dinv);

    auto layer = [&](const float* in, const float* W, const float* b, float* out) {
        gemm_wmma_kernel    <<<N_NODES / 16, 256, 0, stream>>>(in, W, xw);
        selfloop_bias_kernel<<<(N_NODES * DIM) / B, B, 0, stream>>>(xw, dinv, b, out);
        edge_scatter_kernel <<<(N_EDGES * 32) / B, B, 0, stream>>>(src, dst, dinv, xw, out);
    };
    layer(x,  W1, b1, h1);
    layer(h1, W2, b2, h2);
    layer(h2, W3, b3, h3);

    // pooling
    pool_init_kernel   <<<(N_GRAPHS * H3DIM + B - 1) / B, B, 0, stream>>>(pooled, cnt);
    cnt_kernel         <<<(N_NODES + B - 1) / B, B, 0, stream>>>(batch, cnt);
    pool_scatter_kernel<<<(N_NODES * DIM) / B, B, 0, stream>>>(batch, h1, h2, h3, pooled);
    mean_kernel        <<<(N_GRAPHS * H3DIM + B - 1) / B, B, 0, stream>>>(pooled, cnt);

    // head
    lin1_kernel      <<<(N_GRAPHS * H3DIM + B - 1) / B, B, 0, stream>>>(pooled, lin1_W, lin1_b, hmid);
    lin2_kernel      <<<(N_GRAPHS * OUT_DIM + B - 1) / B, B, 0, stream>>>(hmid, lin2_W, lin2_b, (float*)d_out);
    logsoftmax_kernel<<<1, 64, 0, stream>>>((float*)d_out);
}